// Cheb_GCNN_46755013984833
// MI455X (gfx1250) — compile-verified
//
#include <hip/hip_runtime.h>
#include <hip/hip_bf16.h>

// ---------------------------------------------------------------------------
// Cheb-GCNN for MI455X (gfx1250).
//   Dense GEMMs -> V_WMMA_F32_16X16X4_F32. Weights staged in LDS *transposed*
//   (sBT[n][k], k-stride 68 words -> conflict-free ds_load_b64 of each K-pair,
//   zero-padded to K=64 so the WMMA loop is guard-free). All three Chebyshev
//   terms fused into one kernel with compile-time bias+ReLU epilogue.
//   Sparse propagates -> edge-parallel float4 gathers + global_atomic_add_f32
//   scatters; the whole working set fits in the 192MB L2.
// ---------------------------------------------------------------------------

typedef float v2f __attribute__((ext_vector_type(2)));
typedef float v8f __attribute__((ext_vector_type(8)));

#define TPB 256

__device__ __forceinline__ void gAtomicAddF32(float* p, float v) {
  // Lowers to global_atomic_add_f32 (no CAS loop).
  unsafeAtomicAdd(p, v);
}

// ---------------- elementwise helpers ----------------

__global__ void k_zero(float* __restrict__ p, long long n) {
  long long i = (long long)blockIdx.x * blockDim.x + threadIdx.x;
  long long s = (long long)gridDim.x * blockDim.x;
  for (; i < n; i += s) p[i] = 0.0f;
}

// X64[node][f] = (f < F) ? x[node][f] : 0   (stride 50 -> stride 64, 256B rows)
__global__ void k_pad_copy(const float* __restrict__ x, float* __restrict__ X64,
                           int N, int F) {
  long long n = (long long)N * 64;
  long long i = (long long)blockIdx.x * blockDim.x + threadIdx.x;
  long long s = (long long)gridDim.x * blockDim.x;
  for (; i < n; i += s) {
    int node = (int)(i >> 6);
    int f    = (int)(i & 63);
    X64[i] = (f < F) ? x[(long long)node * F + f] : 0.0f;
  }
}

// deg[row[e]] += w[e]
__global__ void k_deg(const int* __restrict__ row, const float* __restrict__ w,
                      float* __restrict__ deg, int E) {
  int i = blockIdx.x * blockDim.x + threadIdx.x;
  int s = gridDim.x * blockDim.x;
  for (; i < E; i += s) gAtomicAddF32(&deg[row[i]], w[i]);
}

// in place: deg -> deg>0 ? rsqrt(deg) : 0
__global__ void k_dis(float* __restrict__ deg, int N) {
  int i = blockIdx.x * blockDim.x + threadIdx.x;
  int s = gridDim.x * blockDim.x;
  for (; i < N; i += s) {
    float d = deg[i];
    deg[i] = (d > 0.0f) ? rsqrtf(d) : 0.0f;
  }
}

// norm[e] = -dis[row]*w*dis[col]
__global__ void k_norm(const int* __restrict__ row, const int* __restrict__ col,
                       const float* __restrict__ w, const float* __restrict__ dis,
                       float* __restrict__ norm, int E) {
  int i = blockIdx.x * blockDim.x + threadIdx.x;
  int s = gridDim.x * blockDim.x;
  for (; i < E; i += s) norm[i] = -dis[row[i]] * w[i] * dis[col[i]];
}

// OUT[row[e]][:] += norm[e] * X[col[e]][:]   (stride-64 rows, nV4 float4 chunks)
__global__ void k_propagate(const float* __restrict__ X, const int* __restrict__ row,
                            const int* __restrict__ col, const float* __restrict__ norm,
                            float* __restrict__ OUT, int E, int nV4) {
  int e = blockIdx.x * blockDim.x + threadIdx.x;
  int s = gridDim.x * blockDim.x;
  for (; e < E; e += s) {
    int r = row[e];
    int c = col[e];
    float w = norm[e];
    const float4* __restrict__ src = (const float4*)(X + (long long)c * 64);
    float* __restrict__ dst = OUT + (long long)r * 64;
    for (int v = 0; v < nV4; ++v) {
      float4 x4 = src[v];
      gAtomicAddF32(dst + 4 * v + 0, w * x4.x);
      gAtomicAddF32(dst + 4 * v + 1, w * x4.y);
      gAtomicAddF32(dst + 4 * v + 2, w * x4.z);
      gAtomicAddF32(dst + 4 * v + 3, w * x4.w);
    }
  }
}

// T2 = 2*T2 - T0
__global__ void k_cheb2(float* __restrict__ T2, const float* __restrict__ T0,
                        long long n) {
  long long i = (long long)blockIdx.x * blockDim.x + threadIdx.x;
  long long s = (long long)gridDim.x * blockDim.x;
  for (; i < n; i += s) T2[i] = 2.0f * T2[i] - T0[i];
}

// ---------------- fused fp32 WMMA GEMM ----------------
// OUT[M x COUT] = relu?( sum_t A_t[M x F] @ W_t + bias ), A row stride = 64,
// A columns beyond F are zero-padded. Weights staged in LDS transposed:
// sBT[t][n][k] with k-stride 68 words:
//   - K-pair (k0, k0+1) contiguous -> one ds_load_b64 straight into the WMMA
//     B operand pair (no register shuffles),
//   - bank = (4*ln + k0 + 2*hi) mod 64 -> all 32 lanes hit distinct banks,
//   - zero-padded to K=64 -> guard-free inner loop.
// One wave computes one 16-row tile across all 16-col tiles;
// V_WMMA_F32_16X16X4_F32 per K-step of 4, KMAX/4 steps (KMAX = 52 for F=50, 64 else).
//   A layout : lane l (hi=l>>4, ln=l&15) holds A[m0+ln][k0], A[m0+ln][k0+1],
//              k0 = kk + 2*hi.
//   B layout : lane l holds B[k0][n0+ln], B[k0+1][n0+ln].
//   C layout : vgpr r, lanes 0-15 -> M=m0+r, lanes 16-31 -> M=m0+r+8.
template <int F, int KMAX, int COUT, int NTERMS, bool TRANSB, bool RELU>
__global__ void k_gemm_wmma(const float* __restrict__ A0, const float* __restrict__ A1,
                            const float* __restrict__ A2, const float* __restrict__ W,
                            const float* __restrict__ bias, float* __restrict__ OUT,
                            int M) {
  constexpr int KS = 68;  // k-row stride in words (even, bank-spreading)
  __shared__ float sBT[NTERMS][COUT][KS];
  // Cooperative stage (transposed, zero-padded K to 64).
  for (int idx = threadIdx.x; idx < NTERMS * COUT * 64; idx += blockDim.x) {
    int t   = idx / (COUT * 64);
    int rem = idx - t * COUT * 64;
    int n   = rem >> 6;
    int k   = rem & 63;
    float v = 0.0f;
    if (k < F) v = TRANSB ? W[n * F + k] : W[t * F * COUT + k * COUT + n];
    sBT[t][n][k] = v;
  }
  __syncthreads();

  const int wavesPerBlock = blockDim.x >> 5;
  int wave = blockIdx.x * wavesPerBlock + (threadIdx.x >> 5);
  int m0 = wave << 4;
  if (m0 + 16 > M) return;  // wave-uniform; EXEC stays all-ones for WMMA
  int lane = threadIdx.x & 31;
  int ln = lane & 15;
  int hi = lane >> 4;

  const float* Arows[3];
  Arows[0] = A0 + (long long)(m0 + ln) * 64 + 2 * hi;
  if (NTERMS > 1) Arows[1] = A1 + (long long)(m0 + ln) * 64 + 2 * hi;
  if (NTERMS > 2) Arows[2] = A2 + (long long)(m0 + ln) * 64 + 2 * hi;

#pragma unroll
  for (int nt = 0; nt < COUT / 16; ++nt) {
    int n = (nt << 4) + ln;
    const float* __restrict__ Bcol = &sBT[0][n][2 * hi];  // + t*COUT*KS + kk
    float bv = bias[n];
    v8f c;
#pragma unroll
    for (int r = 0; r < 8; ++r) c[r] = bv;
#pragma unroll
    for (int t = 0; t < NTERMS; ++t) {
      const float* __restrict__ Arow = Arows[t];
      const float* __restrict__ Bt = Bcol + t * COUT * KS;
#pragma unroll
      for (int kk = 0; kk < KMAX; kk += 4) {
        v2f a = *(const v2f*)(Arow + kk);  // global_load_b64 (padded zeros ok)
        v2f b = *(const v2f*)(Bt + kk);    // ds_load_b64, conflict-free
        c = __builtin_amdgcn_wmma_f32_16x16x4_f32(false, a, false, b, (short)0,
                                                  c, false, false);
      }
    }
#pragma unroll
    for (int r = 0; r < 8; ++r) {
      float v = c[r];
      if (RELU) v = fmaxf(v, 0.0f);
      OUT[(long long)(m0 + r + 8 * hi) * COUT + n] = v;
    }
  }
}

// ---------------- BatchNorm (training-mode batch stats) ----------------

__global__ void k_bn_stats(const float* __restrict__ H, float* __restrict__ stats,
                           long long n) {
  __shared__ float sS[64];
  __shared__ float sQ[64];
  if (threadIdx.x < 64) { sS[threadIdx.x] = 0.0f; sQ[threadIdx.x] = 0.0f; }
  __syncthreads();
  long long i = (long long)blockIdx.x * blockDim.x + threadIdx.x;
  long long s = (long long)gridDim.x * blockDim.x;
  for (; i < n; i += s) {
    float v = H[i];
    int c = (int)(i & 63);
    atomicAdd(&sS[c], v);        // ds_add_f32
    atomicAdd(&sQ[c], v * v);
  }
  __syncthreads();
  if (threadIdx.x < 64) {
    gAtomicAddF32(&stats[threadIdx.x], sS[threadIdx.x]);
    gAtomicAddF32(&stats[64 + threadIdx.x], sQ[threadIdx.x]);
  }
}

__global__ void k_bn_final(const float* __restrict__ stats,
                           const float* __restrict__ gamma,
                           const float* __restrict__ beta,
                           float* __restrict__ params, float invN) {
  int c = threadIdx.x;
  if (c < 64) {
    float mu  = stats[c] * invN;
    float var = stats[64 + c] * invN - mu * mu;
    float sc  = gamma[c] * rsqrtf(var + 1e-5f);
    params[c]      = sc;
    params[64 + c] = beta[c] - mu * sc;
  }
}

__global__ void k_bn_apply(float* __restrict__ H, const float* __restrict__ params,
                           long long n) {
  long long i = (long long)blockIdx.x * blockDim.x + threadIdx.x;
  long long s = (long long)gridDim.x * blockDim.x;
  for (; i < n; i += s) {
    int c = (int)(i & 63);
    H[i] = H[i] * params[c] + params[64 + c];
  }
}

// ---------------- launch ----------------

extern "C" void kernel_launch(void* const* d_in, const int* in_sizes, int n_in,
                              void* d_out, int out_size, void* d_ws, size_t ws_size,
                              hipStream_t stream) {
  const float* x     = (const float*)d_in[0];
  const int*   ei    = (const int*)d_in[1];   // [2,E] row-major: row then col
  const float* ew    = (const float*)d_in[2];
  const float* W1    = (const float*)d_in[3]; // (3,50,64)
  const float* b1    = (const float*)d_in[4];
  const float* W2    = (const float*)d_in[5]; // (3,64,64)
  const float* b2    = (const float*)d_in[6];
  const float* gamma = (const float*)d_in[7];
  const float* beta  = (const float*)d_in[8];
  const float* linW  = (const float*)d_in[9]; // (16,64)
  const float* linb  = (const float*)d_in[10];

  const int IN_F = 50;
  const int N = in_sizes[0] / IN_F;
  const int E = in_sizes[2];
  const int* rowIdx = ei;
  const int* colIdx = ei + E;

  float* ws   = (float*)d_ws;
  float* norm = ws;                         // E
  float* deg  = norm + E;                   // N (then holds dis)
  float* X64  = deg + N;                    // N*64, padded x; reused as H2
  float* T1   = X64 + (long long)N * 64;    // N*64
  float* T2   = T1 + (long long)N * 64;     // N*64
  float* H1   = T2 + (long long)N * 64;     // N*64
  float* bnstats  = H1 + (long long)N * 64; // 128
  float* bnparams = bnstats + 128;          // 128
  float* H2 = X64;                          // reuse after conv1 finishes with X64
  float* out = (float*)d_out;               // N x 16

  const long long NF = (long long)N * 64;
  const int eBlocks  = (E + TPB - 1) / TPB;
  const int nBlocks  = (N + TPB - 1) / TPB;
  const int ewBlocks = 4096;                // grid-stride elementwise over NF
  const int gemmBlocks = ((N / 16) + 7) / 8; // 8 waves/block, 1 tile-row/wave

  // ---- normalization coefficients ----
  k_zero<<<nBlocks, TPB, 0, stream>>>(deg, N);
  k_deg<<<eBlocks, TPB, 0, stream>>>(rowIdx, ew, deg, E);
  k_dis<<<nBlocks, TPB, 0, stream>>>(deg, N);
  k_norm<<<eBlocks, TPB, 0, stream>>>(rowIdx, colIdx, ew, deg, norm, E);

  // ---- pad x to stride 64 ----
  k_pad_copy<<<ewBlocks, TPB, 0, stream>>>(x, X64, N, IN_F);

  // ---- conv1: T0=X64, T1=L.X64, T2=2L.T1-X64 ; H1 = relu(sum Tk@W1[k] + b1)
  k_zero<<<ewBlocks, TPB, 0, stream>>>(T1, NF);
  k_propagate<<<eBlocks, TPB, 0, stream>>>(X64, rowIdx, colIdx, norm, T1, E, 13);
  k_zero<<<ewBlocks, TPB, 0, stream>>>(T2, NF);
  k_propagate<<<eBlocks, TPB, 0, stream>>>(T1, rowIdx, colIdx, norm, T2, E, 13);
  k_cheb2<<<ewBlocks, TPB, 0, stream>>>(T2, X64, NF);
  k_gemm_wmma<50, 52, 64, 3, false, true>
      <<<gemmBlocks, TPB, 0, stream>>>(X64, T1, T2, W1, b1, H1, N);

  // ---- batchnorm (batch stats, eps=1e-5) ----
  k_zero<<<1, 256, 0, stream>>>(bnstats, 128);
  k_bn_stats<<<ewBlocks, TPB, 0, stream>>>(H1, bnstats, NF);
  k_bn_final<<<1, 64, 0, stream>>>(bnstats, gamma, beta, bnparams, 1.0f / (float)N);
  k_bn_apply<<<ewBlocks, TPB, 0, stream>>>(H1, bnparams, NF);

  // ---- conv2 (X64 dead -> H2 aliases it) ----
  k_zero<<<ewBlocks, TPB, 0, stream>>>(T1, NF);
  k_propagate<<<eBlocks, TPB, 0, stream>>>(H1, rowIdx, colIdx, norm, T1, E, 16);
  k_zero<<<ewBlocks, TPB, 0, stream>>>(T2, NF);
  k_propagate<<<eBlocks, TPB, 0, stream>>>(T1, rowIdx, colIdx, norm, T2, E, 16);
  k_cheb2<<<ewBlocks, TPB, 0, stream>>>(T2, H1, NF);
  k_gemm_wmma<64, 64, 64, 3, false, true>
      <<<gemmBlocks, TPB, 0, stream>>>(H1, T1, T2, W2, b2, H2, N);

  // ---- final linear: out = H2 @ linW^T + linb  (B[k][n] = linW[n*64+k]) ----
  k_gemm_wmma<64, 64, 16, 1, true, false>
      <<<gemmBlocks, TPB, 0, stream>>>(H2, H2, H2, linW, linb, out, N);
}